// IntermediateKeypointPredictor_64974265254058
// MI455X (gfx1250) — compile-verified
//
#include <hip/hip_runtime.h>

typedef __attribute__((ext_vector_type(2))) float v2f;
typedef __attribute__((ext_vector_type(8))) float v8f;

#define BATCH 16
#define CH 64
#define HH 128
#define WW 128
#define HWP (HH * WW)          // 16384
#define LATENT 500
#define NA 10
#define LT_PER_BLOCK 32        // 2 WMMA M-tiles of 16 rows
#define BLOCKS_PER_BATCH 16    // 16*32 = 512 >= 500
#define WAVES 8
#define ROWS_PER_WAVE 16       // image rows owned by each wave (8*16 = 128)
#define PHASES 8               // 8 column phases of 16 columns (8*16 = 128)
#define K_LOG2E 72.13475204444817f   // 50 * log2(e): folded into A fragments
#define COORD_STEP (2.0f / 127.0f)

__device__ __forceinline__ float exp2_fast(float x) { return __builtin_amdgcn_exp2f(x); }

// Merge two softmax partial states (all in exp2 domain).
__device__ __forceinline__ void online_merge(float& m, float& s, float& sx, float& sy,
                                             float m2, float s2, float sx2, float sy2) {
  float mn = fmaxf(m, m2);
  float c1 = exp2_fast(m - mn);
  float c2 = exp2_fast(m2 - mn);
  s  = fmaf(s,  c1, s2  * c2);
  sx = fmaf(sx, c1, sx2 * c2);
  sy = fmaf(sy, c1, sy2 * c2);
  m  = mn;
}

__global__ __launch_bounds__(256)
void softargmax_stage1(const float* __restrict__ x,      // [B, C, H, W]
                       const float* __restrict__ Wi,     // [LATENT, C]
                       float* __restrict__ out) {        // d_out base
  const int b    = blockIdx.x / BLOCKS_PER_BATCH;
  const int lblk = blockIdx.x % BLOCKS_PER_BATCH;
  const int l0   = lblk * LT_PER_BLOCK;
  const int tid  = threadIdx.x;
  const int wave = tid >> 5;
  const int lane = tid & 31;
  const int n    = lane & 15;         // M row (A) / N col (B,D)
  const int kb   = (lane >> 4) << 1;  // K offset selected by lane-half

  // Resident A fragments, pre-scaled by 50*log2(e) so WMMA output is exp2-domain.
  v2f a[2][16];
#pragma unroll
  for (int t = 0; t < 2; ++t) {
    int l = l0 + t * 16 + n;
#pragma unroll
    for (int kk = 0; kk < 16; ++kk) {
      v2f av; av.x = 0.0f; av.y = 0.0f;
      if (l < LATENT) {
        const float* wp = Wi + l * CH + kk * 4 + kb;
        av.x = wp[0] * K_LOG2E;
        av.y = wp[1] * K_LOG2E;
      }
      a[t][kk] = av;
    }
  }

  const float* xb = x + (size_t)b * CH * HWP;
  const int rowbase = wave * ROWS_PER_WAVE;

  // Global softmax state per lane: 2 M-tiles x 8 rows.
  float gm[2][8], gS[2][8], gSx[2][8], gSy[2][8];
#pragma unroll
  for (int t = 0; t < 2; ++t)
#pragma unroll
    for (int v = 0; v < 8; ++v) {
      gm[t][v] = -1e30f; gS[t][v] = 0.0f; gSx[t][v] = 0.0f; gSy[t][v] = 0.0f;
    }

#pragma unroll 1
  for (int phase = 0; phase < PHASES; ++phase) {
    const int colbase = phase * 16;
    const int col = colbase + n;                       // this lane's image column
    const float xc = fmaf((float)col, COORD_STEP, -1.0f);  // lane-constant this phase

    // Phase-local state: only (m, S, Sy) needed (xc factored out).
    float pm[2][8], pS[2][8], pSy[2][8];
#pragma unroll
    for (int t = 0; t < 2; ++t)
#pragma unroll
      for (int v = 0; v < 8; ++v) { pm[t][v] = -1e30f; pS[t][v] = 0.0f; pSy[t][v] = 0.0f; }

#pragma unroll 1
    for (int ri = 0; ri < ROWS_PER_WAVE; ++ri) {
      const int row = rowbase + ri;
      const int p = row * WW + col;                    // lane's position this tile

      v8f acc0 = {0.f,0.f,0.f,0.f,0.f,0.f,0.f,0.f};
      v8f acc1 = {0.f,0.f,0.f,0.f,0.f,0.f,0.f,0.f};
#pragma unroll
      for (int kk = 0; kk < 16; ++kk) {
        v2f bf;
        const float* bp = xb + (size_t)(kk * 4 + kb) * HWP + p;
        bf.x = bp[0];
        bf.y = bp[HWP];
        acc0 = __builtin_amdgcn_wmma_f32_16x16x4_f32(false, a[0][kk], false, bf,
                                                     (short)0, acc0, false, false);
        acc1 = __builtin_amdgcn_wmma_f32_16x16x4_f32(false, a[1][kk], false, bf,
                                                     (short)0, acc1, false, false);
      }

      const float yc = fmaf((float)row, COORD_STEP, -1.0f);  // scalar for the tile

#pragma unroll
      for (int v = 0; v < 8; ++v) {
        {
          float val = acc0[v];                         // already 50*log2e scaled
          float mn = fmaxf(pm[0][v], val);
          float c = exp2_fast(pm[0][v] - mn);
          float e = exp2_fast(val - mn);
          pS[0][v]  = fmaf(pS[0][v],  c, e);
          pSy[0][v] = fmaf(pSy[0][v], c, e * yc);
          pm[0][v] = mn;
        }
        {
          float val = acc1[v];
          float mn = fmaxf(pm[1][v], val);
          float c = exp2_fast(pm[1][v] - mn);
          float e = exp2_fast(val - mn);
          pS[1][v]  = fmaf(pS[1][v],  c, e);
          pSy[1][v] = fmaf(pSy[1][v], c, e * yc);
          pm[1][v] = mn;
        }
      }
    }

    // Fold phase partials into global state; Sx contribution = xc * S_phase.
#pragma unroll
    for (int t = 0; t < 2; ++t)
#pragma unroll
      for (int v = 0; v < 8; ++v) {
        float mn = fmaxf(gm[t][v], pm[t][v]);
        float cg = exp2_fast(gm[t][v] - mn);
        float cp = exp2_fast(pm[t][v] - mn);
        float ep = pS[t][v] * cp;
        gS[t][v]  = fmaf(gS[t][v],  cg, ep);
        gSx[t][v] = fmaf(gSx[t][v], cg, ep * xc);
        gSy[t][v] = fmaf(gSy[t][v], cg, pSy[t][v] * cp);
        gm[t][v] = mn;
      }
  }

  // Reduce the 16 columns held by each 16-lane half (butterfly stays within half).
  __shared__ float lds[WAVES][LT_PER_BLOCK][4];
#pragma unroll
  for (int t = 0; t < 2; ++t) {
#pragma unroll
    for (int v = 0; v < 8; ++v) {
      float mm = gm[t][v], ss = gS[t][v], sxx = gSx[t][v], syy = gSy[t][v];
#pragma unroll
      for (int mask = 1; mask <= 8; mask <<= 1) {
        float m2  = __shfl_xor(mm,  mask, 32);
        float s2  = __shfl_xor(ss,  mask, 32);
        float sx2 = __shfl_xor(sxx, mask, 32);
        float sy2 = __shfl_xor(syy, mask, 32);
        online_merge(mm, ss, sxx, syy, m2, s2, sx2, sy2);
      }
      if ((lane & 15) == 0) {
        int r = t * 16 + (lane >> 4) * 8 + v;   // row within this block's 32 rows
        lds[wave][r][0] = mm;
        lds[wave][r][1] = ss;
        lds[wave][r][2] = sxx;
        lds[wave][r][3] = syy;
      }
    }
  }
  __syncthreads();

  if (tid < LT_PER_BLOCK) {
    float mm  = lds[0][tid][0];
    float ss  = lds[0][tid][1];
    float sxx = lds[0][tid][2];
    float syy = lds[0][tid][3];
#pragma unroll
    for (int w2 = 1; w2 < WAVES; ++w2)
      online_merge(mm, ss, sxx, syy,
                   lds[w2][tid][0], lds[w2][tid][1], lds[w2][tid][2], lds[w2][tid][3]);
    int l = l0 + tid;
    if (l < LATENT) {
      float inv = 1.0f / ss;                 // mass == S (division by mass folded in)
      float* inter = out + BATCH * NA * 2;   // intermediate after pred in d_out
      inter[(size_t)b * (LATENT * 2) + l * 2 + 0] = sxx * inv;
      inter[(size_t)b * (LATENT * 2) + l * 2 + 1] = syy * inv;
    }
  }
}

// pred[b,a,o] = sum_n feat[b,a,n] * W_reg[a,o,n]   (feat = intermediate reshaped)
__global__ void reg_stage2(const float* __restrict__ inter,  // [B, LATENT, 2] flat
                           const float* __restrict__ Wr,     // [NA, 2, 100]
                           float* __restrict__ pred) {       // [B, NA, 2]
  int idx = blockIdx.x * blockDim.x + threadIdx.x;
  if (idx >= BATCH * NA * 2) return;
  int o = idx & 1;
  int a = (idx >> 1) % NA;
  int b = idx / (NA * 2);
  const float* f = inter + (size_t)b * (LATENT * 2) + a * 100;
  const float* w = Wr + a * 200 + o * 100;
  float acc = 0.0f;
#pragma unroll 4
  for (int i = 0; i < 100; ++i) acc = fmaf(f[i], w[i], acc);
  pred[idx] = acc;
}

extern "C" void kernel_launch(void* const* d_in, const int* in_sizes, int n_in,
                              void* d_out, int out_size, void* d_ws, size_t ws_size,
                              hipStream_t stream) {
  (void)in_sizes; (void)n_in; (void)out_size; (void)d_ws; (void)ws_size;
  const float* x  = (const float*)d_in[0];   // [16,64,128,128]
  const float* Wi = (const float*)d_in[1];   // [500,64]
  const float* Wr = (const float*)d_in[2];   // [10,2,100]
  float* out = (float*)d_out;                // pred (320) then intermediate (16000)

  softargmax_stage1<<<BATCH * BLOCKS_PER_BATCH, 256, 0, stream>>>(x, Wi, out);
  reg_stage2<<<1, BATCH * NA * 2, 0, stream>>>(out + BATCH * NA * 2, Wr, out);
}